// MyHuberLoss_4097398800619
// MI455X (gfx1250) — compile-verified
//
#include <hip/hip_runtime.h>

// ---------------------------------------------------------------------------
// MyHuberLoss for MI455X (gfx1250, wave32).
// Phase 1: grid-stride streaming kernel; 21 f32 accumulators per lane
//          (Sx[6], Sxx[6], Sxy[6], Sy, Syy, Sloss), coalesced b64 loads,
//          hardware v_exp/v_sqrt/v_rcp for tanh^2 and Huber, wave32 shfl_xor
//          + LDS tree -> one 21-float partial per block (padded to 32).
// Phase 2: one wave32; V_WMMA_F32_16X16X4_F32 with B=ones sums the block
//          partials on the matrix pipe (16 quantities x 4 blocks per wmma,
//          accumulated in the f32 C registers), then lane 0 computes the
//          Pearson terms, penalty, and final scalar loss.
// Deterministic: fixed partition, no atomics, every d_ws slot rewritten.
// ---------------------------------------------------------------------------

#define NBLOCKS 512
#define TPB     512
#define NWAVES  (TPB / 32)
#define NQ      21
#define NQPAD   32

typedef __attribute__((ext_vector_type(2))) float v2f;
typedef __attribute__((ext_vector_type(8))) float v8f;

__device__ __forceinline__ float wave_sum32(float v) {
#pragma unroll
  for (int off = 16; off > 0; off >>= 1)
    v += __shfl_xor(v, off, 32);
  return v;
}

__global__ void huber_partials_kernel(const float* __restrict__ outp,
                                      const float* __restrict__ lab,
                                      float* __restrict__ partial, int n) {
  float acc[NQ];
#pragma unroll
  for (int i = 0; i < NQ; ++i) acc[i] = 0.f;

  const int tid    = blockIdx.x * blockDim.x + threadIdx.x;
  const int stride = gridDim.x * blockDim.x;

  for (int r = tid; r < n; r += stride) {
    const float y = lab[r];
    const float* row = outp + 6ll * (long long)r;   // rows are 8B aligned
    v2f a0 = *(const v2f*)(row);
    v2f a1 = *(const v2f*)(row + 2);
    v2f a2 = *(const v2f*)(row + 4);
    float o[6] = {a0.x, a0.y, a1.x, a1.y, a2.x, a2.y};

    acc[18] += y;        // Sy
    acc[19] += y * y;    // Syy

    float rowloss = 0.f;
#pragma unroll
    for (int i = 0; i < 6; ++i) {
      const float x = o[i];
      acc[i]      += x;        // Sx
      acc[6 + i]  += x * x;    // Sxx
      acc[12 + i] += x * y;    // Sxy
      const float d = x - y;
      // huber = 4*(sqrt(1 + (d/2)^2) - 1)      (DELTA = 2)
      const float hub = 4.f * (__builtin_amdgcn_sqrtf(__builtin_fmaf(0.25f, d * d, 1.f)) - 1.f);
      float w;
      if (i < 5) {
        // tanh(d)^2 = ((e^{2d}-1)/(e^{2d}+1))^2 : one v_exp + one v_rcp
        const float e  = __expf(2.f * d);
        const float th = (e - 1.f) * __builtin_amdgcn_rcpf(e + 1.f);
        w = 0.2f * th * th;
      } else {
        w = 1.f;
      }
      rowloss += hub * w;
    }
    acc[20] += rowloss;  // Sloss
  }

  // intra-wave reduction (wave32)
#pragma unroll
  for (int i = 0; i < NQ; ++i) acc[i] = wave_sum32(acc[i]);

  __shared__ float lds[NWAVES][NQ];
  const int lane = threadIdx.x & 31;
  const int wave = threadIdx.x >> 5;
  if (lane == 0) {
#pragma unroll
    for (int i = 0; i < NQ; ++i) lds[wave][i] = acc[i];
  }
  __syncthreads();

  // quantity-major store: partial[q * NBLOCKS + block]; pad q 21..31 with 0
  if (threadIdx.x < NQPAD) {
    float s = 0.f;
    if (threadIdx.x < NQ) {
#pragma unroll
      for (int w = 0; w < NWAVES; ++w) s += lds[w][threadIdx.x];
    }
    partial[threadIdx.x * NBLOCKS + blockIdx.x] = s;
  }
}

// One wave32 (32 threads). WMMA f32 16x16x4 with B = ones reduces the block
// partials: D[m,n] += sum_k A[m,k], A[m,k] = partial[q(m)][blk + k].
__global__ void huber_final_kernel(const float* __restrict__ partial,
                                   float* __restrict__ outv, int n) {
  const int lane = threadIdx.x;            // 0..31, EXEC all ones for WMMA
  const int m    = lane & 15;              // A-matrix row (quantity offset)
  const int koff = (lane >> 4) * 2;        // lanes 16-31 hold K=2,3

  __shared__ float sums[NQPAD];

  v2f bones;
  bones.x = 1.f;
  bones.y = 1.f;

#pragma unroll
  for (int qc = 0; qc < 2; ++qc) {
    const int qbase = qc * 16;
    v8f c = {};
    const float* base = partial + (long long)(qbase + m) * NBLOCKS + koff;
    for (int blk = 0; blk < NBLOCKS; blk += 4) {
      v2f a = *(const v2f*)(base + blk);
      c = __builtin_amdgcn_wmma_f32_16x16x4_f32(
          /*neg_a=*/false, a, /*neg_b=*/false, bones,
          /*c_mod=*/(short)0, c, /*reuse_a=*/false, /*reuse_b=*/false);
    }
    // D layout: VGPR r -> M=r (lanes 0-15) / M=r+8 (lanes 16-31), any N.
    if (lane == 0) {
#pragma unroll
      for (int r = 0; r < 8; ++r) sums[qbase + r] = c[r];
    }
    if (lane == 16) {
#pragma unroll
      for (int r = 0; r < 8; ++r) sums[qbase + 8 + r] = c[r];
    }
  }
  __syncthreads();

  if (lane == 0) {
    const float fn = (float)n;
    const float Sy = sums[18], Syy = sums[19];
    const float my = Sy / fn;
    const float vy = Syy - fn * my * my;
    const float ry = __builtin_amdgcn_rcpf(__builtin_amdgcn_sqrtf(vy));
    float p[6];
#pragma unroll
    for (int i = 0; i < 6; ++i) {
      const float Sx = sums[i], Sxx = sums[6 + i], Sxy = sums[12 + i];
      const float mx  = Sx / fn;
      const float vx  = Sxx - fn * mx * mx;
      const float cov = Sxy - fn * mx * my;
      p[i] = cov * __builtin_amdgcn_rcpf(__builtin_amdgcn_sqrtf(vx)) * ry;
    }
    const float loss_old = sums[20] / fn;
    const float s = 5.f - p[0] - p[1] - p[2] - p[3] - p[4];
    float pen = 0.f;
#pragma unroll
    for (int i = 0; i < 5; ++i) pen += (1.f - p[i]) * (1.f - p[i]);
    pen = pen * __builtin_amdgcn_rcpf(s);
    outv[0] = loss_old + pen + (1.f - p[5]);
  }
}

extern "C" void kernel_launch(void* const* d_in, const int* in_sizes, int n_in,
                              void* d_out, int out_size, void* d_ws, size_t ws_size,
                              hipStream_t stream) {
  const float* outp = (const float*)d_in[0];   // (n, 6) f32
  const float* lab  = (const float*)d_in[1];   // (n, 1) f32
  float* dst        = (float*)d_out;           // scalar f32 loss
  float* partial    = (float*)d_ws;            // NQPAD * NBLOCKS floats (64 KB)
  const int n       = in_sizes[1];             // 4,000,000 rows

  huber_partials_kernel<<<NBLOCKS, TPB, 0, stream>>>(outp, lab, partial, n);
  huber_final_kernel<<<1, 32, 0, stream>>>(partial, dst, n);
}